// PhotometricLoss_55190329753812
// MI455X (gfx1250) — compile-verified
//
#include <hip/hip_runtime.h>

typedef __attribute__((ext_vector_type(2))) float v2f;
typedef __attribute__((ext_vector_type(8))) float v8f;

namespace {
constexpr int   Wd = 640, Hd = 480;
constexpr float FU = 320.0f, FV = 320.0f;
constexpr float CU = (Wd - 1) * 0.5f;   // 319.5
constexpr float CV = (Hd - 1) * 0.5f;   // 239.5
constexpr float BL = 0.25f;
constexpr int   HW  = Wd * Hd;          // 307200
constexpr int   CHW = 3 * HW;
constexpr int   QPR = Wd / 4;           // 160 quads per row
constexpr int   QPI = HW / 4;           // 76800 quads per image
constexpr long long NPIX = 32LL * HW;   // 9,830,400
}

__global__ void pl_zero_ws(float* ws) { ws[0] = 0.0f; }

__global__ __launch_bounds__(256) void photometric_loss_kernel(
    const float* __restrict__ rgb_right,
    const float* __restrict__ rgb_left,
    const float* __restrict__ depth,
    float* __restrict__ accum) {
  const int b  = blockIdx.y;
  const int q  = blockIdx.x * 256 + threadIdx.x;   // 0 .. QPI-1, exact fit
  const int v  = q / QPR;
  const int u0 = (q - v * QPR) * 4;

  // ---- streaming loads: b128 per channel per quad ----
  const size_t doff = (size_t)b * HW + (size_t)v * Wd + u0;
  const float4 dep4 = *reinterpret_cast<const float4*>(depth + doff);

  const size_t coff = (size_t)b * CHW + (size_t)v * Wd + u0;
  const float4 rr4 = *reinterpret_cast<const float4*>(rgb_right + coff);
  const float4 rg4 = *reinterpret_cast<const float4*>(rgb_right + coff + HW);
  const float4 rb4 = *reinterpret_cast<const float4*>(rgb_right + coff + 2 * HW);
  const float4 lr4 = *reinterpret_cast<const float4*>(rgb_left  + coff);
  const float4 lg4 = *reinterpret_cast<const float4*>(rgb_left  + coff + HW);
  const float4 lb4 = *reinterpret_cast<const float4*>(rgb_left  + coff + 2 * HW);

  const float depa[4] = {dep4.x, dep4.y, dep4.z, dep4.w};
  const float rra[4]  = {rr4.x, rr4.y, rr4.z, rr4.w};
  const float rga[4]  = {rg4.x, rg4.y, rg4.z, rg4.w};
  const float rba[4]  = {rb4.x, rb4.y, rb4.z, rb4.w};
  const float lra[4]  = {lr4.x, lr4.y, lr4.z, lr4.w};
  const float lga[4]  = {lg4.x, lg4.y, lg4.z, lg4.w};
  const float lba[4]  = {lb4.x, lb4.y, lb4.z, lb4.w};

  const float* left_b = rgb_left + (size_t)b * CHW;
  const float Yc = ((float)v - CV) * (1.0f / FV);   // shared by the quad

  float part = 0.0f;
#pragma unroll
  for (int i = 0; i < 4; ++i) {
    const int   u = u0 + i;
    const float d = depa[i];

    // ---- projection (matches reference) ----
    const float Xc = ((float)u - CU) * (1.0f / FU);
    const float X  = d * Xc + BL;
    const float Y  = d * Yc;
    const float Z  = fmaxf(d, 0.001f);
    const float iZ = 1.0f / Z;
    const float Up = FU * X * iZ + CU;
    const float Vp = FV * Y * iZ + CV;
    float un = 2.0f * Up * (1.0f / (float)(Wd - 1)) - 1.0f;
    float vn = 2.0f * Vp * (1.0f / (float)(Hd - 1)) - 1.0f;
    if (fabsf(un) > 1.0f) un = 2.0f;
    if (fabsf(vn) > 1.0f) vn = 2.0f;

    // ---- grid_sample bilinear, zeros padding, align_corners=False ----
    const float x  = ((un + 1.0f) * (float)Wd - 1.0f) * 0.5f;
    const float y  = ((vn + 1.0f) * (float)Hd - 1.0f) * 0.5f;
    const float x0 = floorf(x), y0 = floorf(y);
    const float wx1 = x - x0, wy1 = y - y0;
    const float wx0 = 1.0f - wx1, wy0 = 1.0f - wy1;

    float cr[4], cg[4], cb[4];
    const float yf[2] = {y0, y0 + 1.0f};
    const float xf[2] = {x0, x0 + 1.0f};
#pragma unroll
    for (int j = 0; j < 2; ++j) {
#pragma unroll
      for (int k = 0; k < 2; ++k) {
        const float yy = yf[j], xx = xf[k];
        const bool valid = (xx >= 0.0f) && (xx < (float)Wd) &&
                           (yy >= 0.0f) && (yy < (float)Hd);
        const int xi = (int)fminf(fmaxf(xx, 0.0f), (float)(Wd - 1));
        const int yi = (int)fminf(fmaxf(yy, 0.0f), (float)(Hd - 1));
        const float m = valid ? 1.0f : 0.0f;
        const float* p = left_b + yi * Wd + xi;
        const int c = j * 2 + k;
        cr[c] = p[0]      * m;
        cg[c] = p[HW]     * m;
        cb[c] = p[2 * HW] * m;
      }
    }
    const float w00 = wy0 * wx0, w01 = wy0 * wx1, w10 = wy1 * wx0, w11 = wy1 * wx1;
    const float wr = cr[0] * w00 + cr[1] * w01 + cr[2] * w10 + cr[3] * w11;
    const float wg = cg[0] * w00 + cg[1] * w01 + cg[2] * w10 + cg[3] * w11;
    const float wb = cb[0] * w00 + cb[1] * w01 + cb[2] * w10 + cb[3] * w11;

    // ---- photometric diff + validity mask ----
    const float diff = fabsf(wr - rra[i]) + fabsf(wg - rga[i]) + fabsf(wb - rba[i]);
    const float lsum = lra[i] + lga[i] + lba[i];
    const float wsum = wr + wg + wb;
    part += ((wsum > 0.0f) && (lsum > 0.0f)) ? diff : 0.0f;
  }

  // ---- wave reduction via V_WMMA_F32_16X16X4_F32 (all-f32, exact) ----
  // A(16x4): lane m VGPR0 = A[m,0]=p[m] (lanes 0-15) / A[m,2]=p[16+m] (lanes 16-31),
  // VGPR1 = 0. B(4x16) = ones (layout-independent). D[m,n] = p[m]+p[16+m] in C-layout;
  // lane-local sum of the 8 D VGPRs leaves half-wave sums on lanes 0-15 / 16-31.
  __shared__ float bsum;
  if (threadIdx.x == 0) bsum = 0.0f;
  __syncthreads();

  v2f a;    a.x = part;   a.y = 0.0f;
  v2f ones; ones.x = 1.0f; ones.y = 1.0f;
  v8f cz = {};
  v8f dm = __builtin_amdgcn_wmma_f32_16x16x4_f32(
      /*neg_a=*/false, a, /*neg_b=*/false, ones,
      /*c_mod=*/(short)0, cz, /*reuse_a=*/false, /*reuse_b=*/false);
  const float half_sum = dm[0] + dm[1] + dm[2] + dm[3] + dm[4] + dm[5] + dm[6] + dm[7];

  if ((threadIdx.x & 15) == 0) atomicAdd(&bsum, half_sum);  // lanes 0 & 16 per wave
  __syncthreads();
  if (threadIdx.x == 0) atomicAdd(accum, bsum);             // one global atomic/block
}

__global__ void pl_finalize(const float* ws, float* out) {
  out[0] = ws[0] * (1.0f / (float)NPIX);
}

extern "C" void kernel_launch(void* const* d_in, const int* in_sizes, int n_in,
                              void* d_out, int out_size, void* d_ws, size_t ws_size,
                              hipStream_t stream) {
  const float* rgb_right = (const float*)d_in[0];
  const float* rgb_left  = (const float*)d_in[1];
  const float* depth     = (const float*)d_in[2];
  float* ws  = (float*)d_ws;
  float* out = (float*)d_out;

  pl_zero_ws<<<1, 1, 0, stream>>>(ws);
  dim3 grid(QPI / 256, 32);   // 300 x 32 blocks, 256 threads, 4 px/thread
  photometric_loss_kernel<<<grid, 256, 0, stream>>>(rgb_right, rgb_left, depth, ws);
  pl_finalize<<<1, 1, 0, stream>>>(ws, out);
}